// VRPPolarEdgeEmbedding_2250562863231
// MI455X (gfx1250) — compile-verified
//
#include <hip/hip_runtime.h>
#include <hip/hip_bf16.h>
#include <math.h>

typedef __attribute__((ext_vector_type(2))) float v2f;
typedef __attribute__((ext_vector_type(8))) float v8f;

#define BB 64
#define NN 500
#define DD 128
#define KK 32
#define EE (KK + NN * KK) /* 16032 */

__device__ __forceinline__ unsigned long long packKey(float c, int j) {
    // c >= 0, so its bit pattern is monotonic as unsigned; index in low bits
    // gives jax.lax.top_k's lowest-index-first tie break.
    return ((unsigned long long)__float_as_uint(c) << 32) | (unsigned int)j;
}

// ---------------- Stage A1: polar angles about the depot ----------------
__global__ void theta_kernel(const float* __restrict__ locs, float* __restrict__ theta) {
    int idx = blockIdx.x * blockDim.x + threadIdx.x;
    if (idx >= BB * NN) return;
    int b = idx / NN;
    const float2* L = (const float2*)locs;
    float2 p  = L[idx];
    float2 p0 = L[b * NN];
    theta[idx] = atan2f(p.y - p0.y, p.x - p0.x);
}

// ------------- Stage A2: per-row top-32 on angle cost (wave32/row) -------------
__global__ void topk_rows_kernel(const float* __restrict__ theta, int* __restrict__ Vbuf) {
    int wave = (blockIdx.x * blockDim.x + threadIdx.x) >> 5;
    int lane = threadIdx.x & 31;
    int b = wave / NN;
    int i = wave - b * NN;
    float ti = theta[b * NN + i];

    unsigned long long key[16];
#pragma unroll
    for (int t = 0; t < 16; ++t) {
        int j = lane + 32 * t;
        if (j < NN && j != i) {
            float c = 1.0f - cosf(ti - theta[b * NN + j]);
            key[t] = packKey(c, j);
        } else {
            key[t] = ~0ull; // +inf (self-loop / out of range)
        }
    }

    int base = b * EE + KK + i * KK;
    for (int r = 0; r < KK; ++r) {
        unsigned long long best = key[0];
#pragma unroll
        for (int t = 1; t < 16; ++t) best = (key[t] < best) ? key[t] : best;
        unsigned long long wbest = best;
#pragma unroll
        for (int off = 16; off > 0; off >>= 1) {
            unsigned long long o = __shfl_xor(wbest, off, 32);
            wbest = (o < wbest) ? o : wbest;
        }
        if (lane == 0) Vbuf[base + r] = (int)(wbest & 0xffffffffull);
#pragma unroll
        for (int t = 0; t < 16; ++t)
            if (key[t] == wbest) key[t] = ~0ull; // remove winner (unique key)
    }
}

// ------------- Stage A3: depot top-32 on distance (wave32/batch) -------------
__global__ void topk_depot_kernel(const float* __restrict__ locs, int* __restrict__ Vbuf) {
    int b = blockIdx.x;
    int lane = threadIdx.x;
    const float2* L = (const float2*)locs;
    float2 p0 = L[b * NN];

    unsigned long long key[16];
#pragma unroll
    for (int t = 0; t < 16; ++t) {
        int j = lane + 32 * t;
        if (j < NN) {
            float2 p = L[b * NN + j];
            float dx = p.x - p0.x, dy = p.y - p0.y;
            key[t] = packKey(sqrtf(dx * dx + dy * dy), j); // j==0 (dist 0) selected first, masked later
        } else {
            key[t] = ~0ull;
        }
    }
    int base = b * EE;
    for (int r = 0; r < KK; ++r) {
        unsigned long long best = key[0];
#pragma unroll
        for (int t = 1; t < 16; ++t) best = (key[t] < best) ? key[t] : best;
        unsigned long long wbest = best;
#pragma unroll
        for (int off = 16; off > 0; off >>= 1) {
            unsigned long long o = __shfl_xor(wbest, off, 32);
            wbest = (o < wbest) ? o : wbest;
        }
        if (lane == 0) Vbuf[base + r] = (int)(wbest & 0xffffffffull);
#pragma unroll
        for (int t = 0; t < 16; ++t)
            if (key[t] == wbest) key[t] = ~0ull;
    }
}

// ------------- edge_index + mask outputs -------------
__global__ void edge_meta_kernel(const int* __restrict__ Vbuf,
                                 float* __restrict__ outEI, float* __restrict__ outM) {
    int idx = blockIdx.x * blockDim.x + threadIdx.x;
    if (idx >= BB * EE) return;
    int b = idx / EE;
    int e = idx - b * EE;
    int v = Vbuf[idx];
    int u = (e < KK) ? 0 : (e - KK) / KK;
    int m = (e < KK) ? (v != 0) : ((u != 0) & (v != 0));
    outEI[(size_t)(b * 2 + 0) * EE + e] = (float)(u * m);
    outEI[(size_t)(b * 2 + 1) * EE + e] = (float)(v * m);
    outM[idx] = (float)m;
}

// ------------- Stage B: edge_emb via V_WMMA_F32_16X16X4_F32 -------------
// One wave computes a 16-edge x 16-feature f32 tile: A = attr (K=2 padded to 4),
// B = W tile, C preloaded with bias. Mask applied per row post-WMMA.
__global__ void edge_emb_wmma_kernel(const float* __restrict__ locs,
                                     const float* __restrict__ theta,
                                     const int* __restrict__ Vbuf,
                                     const float* __restrict__ W,
                                     const float* __restrict__ bias,
                                     float* __restrict__ out) {
    int lane = threadIdx.x & 31;
    int dt   = threadIdx.x >> 5; // feature tile 0..7
    int et   = blockIdx.x;       // edge tile 0..1001
    int b    = blockIdx.y;

    int half = lane >> 4;  // 0: lanes 0-15 (K=0,1) ; 1: lanes 16-31 (K=2,3 -> zero)
    int l15  = lane & 15;

    // Per-edge attr (lanes 16-31 mirror lanes 0-15's addresses; coalesced dup loads)
    int e = et * 16 + l15;
    int v = Vbuf[b * EE + e];
    int u = (e < KK) ? 0 : (e - KK) / KK;
    float tu = theta[b * NN + u];
    float tv = theta[b * NN + v];
    const float2* L = (const float2*)locs;
    float2 pu = L[b * NN + u];
    float2 pv = L[b * NN + v];
    float a1 = 1.0f - cosf(tu - tv);
    float dx = pu.x - pv.x, dy = pu.y - pv.y;
    float ds = sqrtf(dx * dx + dy * dy);

    // A (16x4 f32): lanes 0-15 hold A[m][0],A[m][1]; lanes 16-31 hold A[m][2],A[m][3]=0
    v2f A;
    A.x = half ? 0.0f : a1;
    A.y = half ? 0.0f : ds;

    // B (4x16 f32): rows K=0,1 in lanes 0-15 (col = lane); rows K=2,3 zero
    int n = dt * 16 + l15;
    v2f Bm;
    Bm.x = half ? 0.0f : W[n * 2 + 0];
    Bm.y = half ? 0.0f : W[n * 2 + 1];

    // C: bias depends only on column N = n -> broadcast across the 8 row-VGPRs
    float bv = bias[n];
    v8f C = {bv, bv, bv, bv, bv, bv, bv, bv};

    v8f Dm = __builtin_amdgcn_wmma_f32_16x16x4_f32(
        /*neg_a=*/false, A, /*neg_b=*/false, Bm,
        /*c_mod=*/(short)0, C, /*reuse_a=*/false, /*reuse_b=*/false);

    // Row mask: edge j's mask lives in lanes j and j+16
    int   m  = (e < KK) ? (v != 0) : ((u != 0) & (v != 0));
    float mf = (float)m;

    size_t rowbase = ((size_t)b * EE + (size_t)et * 16) * DD + (size_t)n;
#pragma unroll
    for (int r = 0; r < 8; ++r) {
        int row = half ? (8 + r) : r;          // D-layout: lane<16 -> M=r, lane>=16 -> M=8+r
        float rm = __shfl(mf, row, 32);
        out[rowbase + (size_t)row * DD] = Dm[r] * rm;
    }
}

extern "C" void kernel_launch(void* const* d_in, const int* in_sizes, int n_in,
                              void* d_out, int out_size, void* d_ws, size_t ws_size,
                              hipStream_t stream) {
    (void)in_sizes; (void)n_in; (void)out_size; (void)ws_size;
    const float* locs     = (const float*)d_in[0];
    const float* init_emb = (const float*)d_in[1];
    const float* W        = (const float*)d_in[2];
    const float* bias     = (const float*)d_in[3];
    // d_in[4] = k_sparse (fixed at 32 in the reference)

    const size_t SZ0 = (size_t)BB * NN * DD;   // init_embeddings passthrough
    const size_t SZ1 = (size_t)BB * 2 * EE;    // edge_index
    const size_t SZ2 = (size_t)BB * EE * DD;   // edge_emb

    float* out     = (float*)d_out;
    float* out_ei  = out + SZ0;
    float* out_emb = out + SZ0 + SZ1;
    float* out_m   = out + SZ0 + SZ1 + SZ2;

    float* theta = (float*)d_ws;                                   // BB*NN floats
    int*   Vbuf  = (int*)((char*)d_ws + sizeof(float) * BB * NN);  // BB*EE ints

    // Output 0: passthrough copy (graph-capture-safe async d2d)
    hipMemcpyAsync(d_out, init_emb, SZ0 * sizeof(float), hipMemcpyDeviceToDevice, stream);

    theta_kernel<<<(BB * NN + 255) / 256, 256, 0, stream>>>(locs, theta);
    topk_rows_kernel<<<(BB * NN) / 8, 256, 0, stream>>>(theta, Vbuf);   // 8 waves/block
    topk_depot_kernel<<<BB, 32, 0, stream>>>(locs, Vbuf);
    edge_meta_kernel<<<(BB * EE + 255) / 256, 256, 0, stream>>>(Vbuf, out_ei, out_m);
    edge_emb_wmma_kernel<<<dim3(EE / 16, BB), 256, 0, stream>>>(locs, theta, Vbuf, W, bias, out_emb);
}